// HyperLoss_11227044512028
// MI455X (gfx1250) — compile-verified
//
#include <hip/hip_runtime.h>
#include <hip/hip_bf16.h>
#include <math.h>

#define NPTS 8192
#define DDIM 64
#define NTILES (NPTS / 16)            // 512 row/col tiles
#define CSPLIT 8                      // waves per row tile (one 256-thread block)
#define JT_PER_WAVE (NTILES / CSPLIT) // 64 column tiles per wave

typedef float v2f __attribute__((ext_vector_type(2)));
typedef float v8f __attribute__((ext_vector_type(8)));

#define EPSF   1e-7f
#define LN2F   0.6931471805599453f

// Pass 0: per-point aux = { sq, 1/(1-sq), label_bits, 0 }; zero the output.
__global__ void hl_aux_kernel(const float* __restrict__ x,
                              const int* __restrict__ labels,
                              float4* __restrict__ aux,
                              float* __restrict__ out) {
    int i = blockIdx.x * blockDim.x + threadIdx.x;
    if (i == 0) out[0] = 0.0f;
    if (i < NPTS) {
        const float4* p = (const float4*)(x + (size_t)i * DDIM);
        float s = 0.0f;
#pragma unroll
        for (int t = 0; t < DDIM / 4; ++t) {
            float4 v = p[t];
            s += v.x * v.x + v.y * v.y + v.z * v.z + v.w * v.w;
        }
        float q = 1.0f / (1.0f - s);   // 1-s >= 0.19 for this data (norm<0.9)
        aux[i] = make_float4(s, q, __int_as_float(labels[i]), 0.0f);
    }
}

// Pass 1: WMMA Gram tiles fused with Poincare distance + streaming sums.
// exp(-arccosh(arg)) == arg - sqrt(arg^2-1): the softmax denominator needs
// no transcendentals; d (log) is needed only for the positive-pair sum.
__global__ __launch_bounds__(256) void hl_tile_kernel(
        const float* __restrict__ x,
        const float4* __restrict__ aux,
        float* __restrict__ part) {
    const int wave = threadIdx.x >> 5;
    const int lane = threadIdx.x & 31;
    const int half = lane >> 4;   // K-parity half for A/B, row-half for C
    const int lq   = lane & 15;
    const int rt    = blockIdx.x;
    const int ibase = rt * 16;

    // ---- Preload A (16x4 f32 per step): lane holds row ibase+lq,
    // K offsets 4*t + 2*half + {0,1}  (ISA 32-bit A 16x4 layout).
    v2f A[16];
    {
        const float* arow = x + (size_t)(ibase + lq) * DDIM + 2 * half;
#pragma unroll
        for (int t = 0; t < 16; ++t)
            A[t] = *(const v2f*)(arow + 4 * t);
    }

    // ---- Row-side constants for this lane's C rows: i = ibase + 8*half + m
    float sqi[8], c2qi[8];
    int   rlbl[8];
#pragma unroll
    for (int m = 0; m < 8; ++m) {
        float4 a = aux[ibase + 8 * half + m];
        sqi[m]  = a.x;
        c2qi[m] = 2.0f * a.y;
        rlbl[m] = __float_as_int(a.z);
    }

    float rsum[8], psum[8], pcnt[8];
#pragma unroll
    for (int m = 0; m < 8; ++m) { rsum[m] = 0.0f; psum[m] = 0.0f; pcnt[m] = 0.0f; }

    const int jt0 = wave * JT_PER_WAVE;
    for (int jt = jt0; jt < jt0 + JT_PER_WAVE; ++jt) {
        const int jbase = jt * 16;

        // B (4x16 f32 per step): lane holds col jbase+lq, same K pattern as A.
        v8f c = {};
        const float* brow = x + (size_t)(jbase + lq) * DDIM + 2 * half;
#pragma unroll
        for (int t = 0; t < 16; ++t) {
            v2f b = *(const v2f*)(brow + 4 * t);
            c = __builtin_amdgcn_wmma_f32_16x16x4_f32(
                    false, A[t], false, b, (short)0, c, false, false);
        }

        const int    j  = jbase + lq;      // this lane's global column
        const float4 aj = aux[j];
        const float  sqj  = aj.x;
        const float  qj   = aj.y;
        const int    lblj = __float_as_int(aj.z);

        float sij[8], cf[8];
#pragma unroll
        for (int m = 0; m < 8; ++m) { sij[m] = sqi[m] + sqj; cf[m] = c2qi[m] * qj; }

#pragma unroll
        for (int m = 0; m < 8; ++m) {
            float sqd = fmaf(-2.0f, c[m], sij[m]);
            sqd       = fmaxf(sqd, 0.0f);
            float arg = fmaf(sqd, cf[m], 1.0f);
            arg       = fmaxf(arg, 1.0f + EPSF);
            float s   = __builtin_amdgcn_sqrtf(fmaf(arg, arg, -1.0f));
            float e   = arg - s;                                // exp(-d)
            float d   = LN2F * __builtin_amdgcn_logf(arg + s);  // arccosh(arg)
            int   i   = ibase + 8 * half + m;
            bool  nd  = (j != i);
            bool  pos = nd & (lblj == rlbl[m]);
            rsum[m] += nd  ? e : 0.0f;
            psum[m] += pos ? d : 0.0f;
            pcnt[m] += pos ? 1.0f : 0.0f;
        }
    }

    // ---- Merge the 16 column classes within each lane half (xor masks <16
    // never cross the half boundary on wave32).
#pragma unroll
    for (int m = 0; m < 8; ++m) {
#pragma unroll
        for (int off = 8; off >= 1; off >>= 1) {
            rsum[m] += __shfl_xor(rsum[m], off);
            psum[m] += __shfl_xor(psum[m], off);
            pcnt[m] += __shfl_xor(pcnt[m], off);
        }
    }

    if (lq == 0) { // lane 0 -> rows 0..7, lane 16 -> rows 8..15
#pragma unroll
        for (int m = 0; m < 8; ++m) {
            int row = 8 * half + m;
            float* p = part + (((size_t)rt * CSPLIT + wave) * 16 + row) * 4;
            p[0] = rsum[m]; p[1] = psum[m]; p[2] = pcnt[m]; p[3] = 0.0f;
        }
    }
}

// Pass 2: merge CSPLIT partials per row, form per-row loss term, reduce.
__global__ void hl_finish_kernel(const float* __restrict__ part,
                                 float* __restrict__ out) {
    int i = blockIdx.x * blockDim.x + threadIdx.x; // row id
    float term = 0.0f;
    if (i < NPTS) {
        int rt = i >> 4, row = i & 15;
        float s = 0.0f, ps = 0.0f, pc = 0.0f;
#pragma unroll
        for (int w = 0; w < CSPLIT; ++w) {
            const float* p = part + (((size_t)rt * CSPLIT + w) * 16 + row) * 4;
            s  += p[0];
            ps += p[1];
            pc += p[2];
        }
        float logA = LN2F * __builtin_amdgcn_logf(s); // logsumexp with max=0
        term = ps / pc + logA;                        // = -p_sum_i / P_i
    }
    __shared__ float red[256];
    red[threadIdx.x] = term;
    __syncthreads();
    for (int off = 128; off > 0; off >>= 1) {
        if ((int)threadIdx.x < off) red[threadIdx.x] += red[threadIdx.x + off];
        __syncthreads();
    }
    if (threadIdx.x == 0) atomicAdd(out, red[0]);
}

extern "C" void kernel_launch(void* const* d_in, const int* in_sizes, int n_in,
                              void* d_out, int out_size, void* d_ws, size_t ws_size,
                              hipStream_t stream) {
    (void)in_sizes; (void)n_in; (void)out_size; (void)ws_size;
    const float* points = (const float*)d_in[0];
    const int*   labels = (const int*)d_in[1];
    float*  out  = (float*)d_out;
    float4* aux  = (float4*)d_ws;                    // 8192 * 16B = 128 KB
    float*  part = (float*)d_ws + 4 * NPTS;          // 512*8*16*4 floats = 1 MB

    hl_aux_kernel<<<(NPTS + 255) / 256, 256, 0, stream>>>(points, labels, aux, out);
    hl_tile_kernel<<<NTILES, 256, 0, stream>>>(points, aux, part);
    hl_finish_kernel<<<(NPTS + 255) / 256, 256, 0, stream>>>(part, out);
}